// VQLayer_86320252715233
// MI455X (gfx1250) — compile-verified
//
#include <hip/hip_runtime.h>
#include <math.h>

typedef __attribute__((ext_vector_type(2))) float v2f;
typedef __attribute__((ext_vector_type(4))) float v4f;
typedef __attribute__((ext_vector_type(8))) float v8f;

#define B_N 32768
#define K_N 1024
#define D_N 512
#define EPSF 1e-7f
#define KLW 1.0f
#define ENTW 0.01f

// ---------------- zero init of accumulators ----------------
__global__ void k_zero(float* __restrict__ p, int n) {
    int i = blockIdx.x * blockDim.x + threadIdx.x;
    if (i < n) p[i] = 0.0f;
}

// ---------------- p2[k] = sum_d protos[k][d]^2 ----------------
__global__ __launch_bounds__(256) void k_p2(const float* __restrict__ protos,
                                            float* __restrict__ p2) {
    int wave = threadIdx.x >> 5, lane = threadIdx.x & 31;
    int row = blockIdx.x * 8 + wave;           // grid = 128 blocks -> 1024 rows
    const float* pr = protos + (size_t)row * D_N;
    float s = 0.0f;
    for (int d = lane * 4; d < D_N; d += 128) {
        v4f v = *(const v4f*)(pr + d);
        s += v[0] * v[0] + v[1] * v[1] + v[2] * v[2] + v[3] * v[3];
    }
    #pragma unroll
    for (int off = 16; off; off >>= 1) s += __shfl_xor(s, off, 32);
    if (lane == 0) p2[row] = s;
}

// ---------------- fused GEMM (f32 WMMA) + gumbel log-softmax ----------------
// block = 256 threads (8 waves). Each block: 16 rows x 1024 cols.
// Each wave: 16 rows x 128 cols = 8 WMMA 16x16 accumulators, K-dim loop step 4.
__global__ __launch_bounds__(256) void k_gemm_softmax(
    const float* __restrict__ latents, const float* __restrict__ protos,
    const float* __restrict__ noise,   const float* __restrict__ p2,
    float* __restrict__ colsum_y, float* __restrict__ colsum_lp,
    int* __restrict__ idx_out)
{
    __shared__ float smem[16][K_N];   // 64 KB logits staging
    __shared__ float s_lse[16];

    const int tid  = threadIdx.x;
    const int wave = tid >> 5, lane = tid & 31;
    const int rowBase = blockIdx.x * 16;
    const int colBase = wave * 128;
    const int r  = lane & 15;          // A row / B col within 16-tile
    const int ko = (lane >> 4) << 1;   // K-pair offset: 0 or 2

    const float* Aptr = latents + (size_t)(rowBase + r) * D_N + ko;
    const float* Bptr = protos  + (size_t)(colBase + r) * D_N + ko;

    v8f acc[8];
    #pragma unroll
    for (int t = 0; t < 8; ++t) acc[t] = (v8f)(0.0f);

    for (int d = 0; d < D_N; d += 4) {
        v2f a = *(const v2f*)(Aptr + d);   // A frag: 16x4, contiguous K-pair
        #pragma unroll
        for (int t = 0; t < 8; ++t) {
            v2f b = *(const v2f*)(Bptr + (size_t)t * 16 * D_N + d);  // B frag
            acc[t] = __builtin_amdgcn_wmma_f32_16x16x4_f32(
                false, a, false, b, (short)0, acc[t], false, false);
        }
    }

    // epilogue: logit = 2*dot - p2[col] + gumbel   (||x||^2 cancels in softmax)
    const int mOff = (lane >> 4) * 8;
    #pragma unroll
    for (int t = 0; t < 8; ++t) {
        int col = colBase + t * 16 + r;
        float pp = p2[col];
        #pragma unroll
        for (int i = 0; i < 8; ++i) {
            int m = mOff + i;
            float u = noise[(size_t)(rowBase + m) * K_N + col];
            float g = -__logf(-__logf(u + 1e-20f) + 1e-20f);
            smem[m][col] = 2.0f * acc[t][i] - pp + g;   // TAU == 1
        }
    }
    __syncthreads();

    // per-row: max+argmax, then log-sum-exp (each wave handles 2 rows)
    #pragma unroll
    for (int rr = 0; rr < 2; ++rr) {
        int m = wave * 2 + rr;
        float vmax = -3.4e38f; int vidx = 0;
        for (int n = lane; n < K_N; n += 32) {
            float v = smem[m][n];
            if (v > vmax) { vmax = v; vidx = n; }
        }
        #pragma unroll
        for (int off = 16; off; off >>= 1) {
            float om = __shfl_xor(vmax, off, 32);
            int   oi = __shfl_xor(vidx, off, 32);
            if (om > vmax || (om == vmax && oi < vidx)) { vmax = om; vidx = oi; }
        }
        float s = 0.0f;
        for (int n = lane; n < K_N; n += 32) s += __expf(smem[m][n] - vmax);
        #pragma unroll
        for (int off = 16; off; off >>= 1) s += __shfl_xor(s, off, 32);
        if (lane == 0) {
            s_lse[m] = vmax + __logf(s);
            idx_out[rowBase + m] = vidx;
        }
    }
    __syncthreads();

    // column sums over this block's 16 rows -> global atomics
    for (int n = tid; n < K_N; n += 256) {
        float sy = 0.0f, slp = 0.0f;
        #pragma unroll
        for (int m = 0; m < 16; ++m) {
            float lp = smem[m][n] - s_lse[m];
            slp += lp;
            sy  += __expf(lp);
        }
        atomicAdd(&colsum_y[n], sy);
        atomicAdd(&colsum_lp[n], slp);
    }
}

// ---------------- prior normalization, KL complexity, entropy ----------------
__global__ __launch_bounds__(256) void k_prior(const float* __restrict__ cy,
                                               const float* __restrict__ clp,
                                               float* __restrict__ klent) {
    __shared__ float red[3][8];
    const int tid = threadIdx.x, lane = tid & 31, wave = tid >> 5;
    const float invB = 1.0f / (float)B_N;

    float loc = 0.0f;
    for (int n = tid; n < K_N; n += 256) loc += cy[n] * invB + EPSF;
    #pragma unroll
    for (int off = 16; off; off >>= 1) loc += __shfl_xor(loc, off, 32);
    if (lane == 0) red[0][wave] = loc;
    __syncthreads();
    float tot = 0.0f;
    #pragma unroll
    for (int w = 0; w < 8; ++w) tot += red[0][w];

    float a1 = 0.0f, a2 = 0.0f;   // sum p*log p ; sum p * mean_b(logprob)
    for (int n = tid; n < K_N; n += 256) {
        float p  = (cy[n] * invB + EPSF) / tot;
        float lp = __logf(p);
        a1 += p * lp;
        a2 += p * (clp[n] * invB);
    }
    #pragma unroll
    for (int off = 16; off; off >>= 1) {
        a1 += __shfl_xor(a1, off, 32);
        a2 += __shfl_xor(a2, off, 32);
    }
    if (lane == 0) { red[1][wave] = a1; red[2][wave] = a2; }
    __syncthreads();
    if (tid == 0) {
        float s1 = 0.0f, s2 = 0.0f;
        #pragma unroll
        for (int w = 0; w < 8; ++w) { s1 += red[1][w]; s2 += red[2][w]; }
        float complexity = s1 - s2;   // sum_k p*log p - sum_k p * mean logprob
        float ent = -s1;
        klent[0] = KLW * complexity + ENTW * ent;
    }
}

// ---------------- quantized = protos[idx]; accumulate sum((q-x)^2) ----------------
__global__ __launch_bounds__(256) void k_quant(const float* __restrict__ latents,
                                               const float* __restrict__ protos,
                                               const int* __restrict__ idx,
                                               float* __restrict__ out_q,
                                               float* __restrict__ sqacc) {
    __shared__ float red[8];
    const int tid = threadIdx.x, lane = tid & 31, wave = tid >> 5;
    const size_t nvec = (size_t)B_N * D_N / 4;
    const size_t stride = (size_t)gridDim.x * blockDim.x;
    float loc = 0.0f;
    for (size_t e = (size_t)blockIdx.x * blockDim.x + tid; e < nvec; e += stride) {
        size_t elem = e * 4;
        int b = (int)(elem >> 9);        // /D_N
        int d = (int)(elem & (D_N - 1));
        int k = idx[b];
        v4f q = *(const v4f*)(protos + (size_t)k * D_N + d);
        v4f x = *(const v4f*)(latents + elem);
        *(v4f*)(out_q + elem) = q;
        #pragma unroll
        for (int j = 0; j < 4; ++j) { float df = q[j] - x[j]; loc += df * df; }
    }
    #pragma unroll
    for (int off = 16; off; off >>= 1) loc += __shfl_xor(loc, off, 32);
    if (lane == 0) red[wave] = loc;
    __syncthreads();
    if (tid == 0) {
        float s = 0.0f;
        #pragma unroll
        for (int w = 0; w < 8; ++w) s += red[w];
        atomicAdd(sqacc, s);
    }
}

// ---------------- final scalar combine ----------------
__global__ void k_final(const float* __restrict__ klent,
                        const float* __restrict__ sq,
                        float* __restrict__ out_loss) {
    // embedding_loss + 0.25*commitment_loss == 1.25 * mean((q-x)^2)
    out_loss[0] = klent[0] + 1.25f * sq[0] / (float)((size_t)B_N * D_N);
}

extern "C" void kernel_launch(void* const* d_in, const int* in_sizes, int n_in,
                              void* d_out, int out_size, void* d_ws, size_t ws_size,
                              hipStream_t stream) {
    const float* latents = (const float*)d_in[0];   // [B, D]
    const float* protos  = (const float*)d_in[1];   // [K, D]
    const float* noise   = (const float*)d_in[2];   // [B, K]
    float* out = (float*)d_out;                     // [B*D] quantized + [1] loss

    float* wsf       = (float*)d_ws;
    float* colsum_y  = wsf;                 // [K]
    float* colsum_lp = wsf + K_N;           // [K]
    float* klent     = wsf + 2 * K_N;       // [1]
    float* sqacc     = wsf + 2 * K_N + 1;   // [1]
    float* p2        = wsf + 2 * K_N + 2;   // [K]
    int*   idx       = (int*)(wsf + 4096);  // [B]

    const int nz = 2 * K_N + 2;
    k_zero<<<(nz + 255) / 256, 256, 0, stream>>>(wsf, nz);
    k_p2<<<K_N / 8, 256, 0, stream>>>(protos, p2);
    k_gemm_softmax<<<B_N / 16, 256, 0, stream>>>(latents, protos, noise, p2,
                                                 colsum_y, colsum_lp, idx);
    k_prior<<<1, 256, 0, stream>>>(colsum_y, colsum_lp, klent);
    k_quant<<<4096, 256, 0, stream>>>(latents, protos, idx, out, sqacc);
    k_final<<<1, 1, 0, stream>>>(klent, sqacc, out + (size_t)B_N * D_N);
}